// GlassBlur_29497835389451
// MI455X (gfx1250) — compile-verified
//
#include <hip/hip_runtime.h>
#include <math.h>

#define HW (1024 * 1024)
#define NW 1022  // inner width = W - 2*MD

// reflect padding (no edge repeat): -1 -> 1, 1024 -> 1022
__device__ __forceinline__ int reflect1(int i) {
    if (i < 0) return -i;
    if (i > 1023) return 2046 - i;
    return i;
}

// Low 32 bits of a generic pointer to __shared__ == LDS byte offset
// (ISA: the shared aperture base has zero low 32 bits).
__device__ __forceinline__ unsigned lds_addr32(const void* p) {
    return (unsigned)(unsigned long long)p;
}

// ---------------------------------------------------------------------------
// Kernel 1: first Gaussian blur (3x3 separable, reflect pad).
// Stages a 34-row x 40-col reflect-mapped tile into LDS with async b128
// loads (CDNA5 async-to-LDS path), then vertical-then-horizontal blur,
// one 1x4 pixel group per thread, b128 stores.
// ---------------------------------------------------------------------------
__global__ __launch_bounds__(256) void k_blur1(const float* __restrict__ img,
                                               float* __restrict__ b1,
                                               float w0, float w1) {
    const int tx = blockIdx.x, ty = blockIdx.y, ch = blockIdx.z;
    const float* src = img + ch * HW;
    float* dst = b1 + ch * HW;

    __shared__ __align__(16) float tile[34 * 40];

    // 16-byte-aligned 40-float column span covering tile cols +/- 1 halo.
    int C0 = tx * 32 - 4;
    C0 = C0 < 0 ? 0 : (C0 > 984 ? 984 : C0);
    const int R0 = ty * 32 - 1;

    // Stage 34 rows x 10 b128-chunks, reflect applied to the source row index.
    for (int chunk = threadIdx.x; chunk < 34 * 10; chunk += 256) {
        const int r = chunk / 10;
        const int c4 = chunk - r * 10;
        const int gr = reflect1(R0 + r);
        const unsigned long long g =
            (unsigned long long)(src + gr * 1024 + C0 + c4 * 4);
        const unsigned laddr = lds_addr32(&tile[r * 40 + c4 * 4]);
        asm volatile("global_load_async_to_lds_b128 %0, %1, off"
                     :: "v"(laddr), "v"(g)
                     : "memory");
    }
    asm volatile("s_wait_asynccnt 0x0" ::: "memory");
    __syncthreads();

    // One 1x4 pixel group per thread: li in 0..31, lj4 in {0,4,...,28}.
    const int li = threadIdx.x >> 3;
    const int lj4 = (threadIdx.x & 7) * 4;
    const int i = ty * 32 + li;
    const int j0 = tx * 32 + lj4;

    // Vertical 3-tap into 6 column sums (matches reference f32 order).
    float v[6];
#pragma unroll
    for (int m = 0; m < 6; ++m) {
        const int cc = reflect1(j0 - 1 + m) - C0;
        v[m] = w0 * tile[li * 40 + cc] + w1 * tile[(li + 1) * 40 + cc] +
               w0 * tile[(li + 2) * 40 + cc];
    }
    // Horizontal 3-tap over the column sums.
    float4 o;
    o.x = w0 * v[0] + w1 * v[1] + w0 * v[2];
    o.y = w0 * v[1] + w1 * v[2] + w0 * v[3];
    o.z = w0 * v[2] + w1 * v[3] + w0 * v[4];
    o.w = w0 * v[3] + w1 * v[4] + w0 * v[5];
    *reinterpret_cast<float4*>(dst + i * 1024 + j0) = o;
}

// ---------------------------------------------------------------------------
// Closed-form parallel replacement for the sequential swap scan:
// trace slot (i,j) through the reversed swap list (ascending (h,w) order).
// Step at (h,w) [valid for 2<=h,w<=1023] swaps (h,w) <-> (h+dy, w+dx),
// dx,dy in {-1,0}; forward index t = (1023-h)*1022 + (1023-w).
// Returns the source pixel index perm(i*1024+j).
// ---------------------------------------------------------------------------
__device__ __forceinline__ int trace_perm(int i, int j,
                                          const int2* __restrict__ dxdy) {
    int cur = -1;  // lexicographic key (h<<12)|w of last applied step
    for (;;) {
        // Candidate A: step at (i,j) — always touches slot (i,j).
        if (i >= 2 && j >= 2) {
            const int key = (i << 12) | j;
            if (key > cur) {
                const int2 d = dxdy[(1023 - i) * NW + (1023 - j)];
                cur = key;
                i += d.y;  // dy
                j += d.x;  // dx
                continue;
            }
        }
        // Candidate B1: step at (i, j+1) with (dy,dx)==(0,-1).
        if (i >= 2 && j + 1 >= 2 && j + 1 <= 1023) {
            const int key = (i << 12) | (j + 1);
            if (key > cur) {
                const int2 d = dxdy[(1023 - i) * NW + (1022 - j)];
                if (d.x == -1 && d.y == 0) { cur = key; j += 1; continue; }
            }
        }
        // Candidate B2: step at (i+1, j) with (dy,dx)==(-1,0).
        if (i + 1 >= 2 && i + 1 <= 1023 && j >= 2) {
            const int key = ((i + 1) << 12) | j;
            if (key > cur) {
                const int2 d = dxdy[(1022 - i) * NW + (1023 - j)];
                if (d.x == 0 && d.y == -1) { cur = key; i += 1; continue; }
            }
        }
        // Candidate B3: step at (i+1, j+1) with (dy,dx)==(-1,-1).
        if (i + 1 >= 2 && i + 1 <= 1023 && j + 1 >= 2 && j + 1 <= 1023) {
            const int key = ((i + 1) << 12) | (j + 1);
            if (key > cur) {
                const int2 d = dxdy[(1022 - i) * NW + (1022 - j)];
                if (d.x == -1 && d.y == -1) { cur = key; i += 1; j += 1; continue; }
            }
        }
        break;
    }
    return i * 1024 + j;
}

// ---------------------------------------------------------------------------
// Kernel 2: fused permutation-gather + second blur + clip.
// Each block builds X = b1[perm(.)] for its 34x34 halo region (3 channels)
// in LDS, then applies the 3x3 blur (vertical-then-horizontal) and clamps.
// ---------------------------------------------------------------------------
__global__ __launch_bounds__(256) void k_glass2(const float* __restrict__ b1,
                                                const int2* __restrict__ dxdy,
                                                float* __restrict__ out,
                                                float w0, float w1) {
    const int tx = blockIdx.x, ty = blockIdx.y;
    __shared__ __align__(16) float X[3][34 * 36];

    for (int p = threadIdx.x; p < 34 * 34; p += 256) {
        const int r = p / 34, c = p - r * 34;
        const int qi = reflect1(ty * 32 - 1 + r);
        const int qj = reflect1(tx * 32 - 1 + c);
        const int s = trace_perm(qi, qj, dxdy);
        X[0][r * 36 + c] = b1[s];
        X[1][r * 36 + c] = b1[HW + s];
        X[2][r * 36 + c] = b1[2 * HW + s];
    }
    __syncthreads();

    // One 1x4 pixel group per thread; X local col for pixel j is lj+1,
    // halo already reflect-mapped at staging time.
    const int li = threadIdx.x >> 3;
    const int lj4 = (threadIdx.x & 7) * 4;
    const int i = ty * 32 + li;
    const int j0 = tx * 32 + lj4;

#pragma unroll
    for (int ch = 0; ch < 3; ++ch) {
        float v[6];
#pragma unroll
        for (int m = 0; m < 6; ++m) {
            const int cc = lj4 + m;  // X cols lj4 .. lj4+5
            v[m] = w0 * X[ch][li * 36 + cc] + w1 * X[ch][(li + 1) * 36 + cc] +
                   w0 * X[ch][(li + 2) * 36 + cc];
        }
        float4 o;
        o.x = w0 * v[0] + w1 * v[1] + w0 * v[2];
        o.y = w0 * v[1] + w1 * v[2] + w0 * v[3];
        o.z = w0 * v[2] + w1 * v[3] + w0 * v[4];
        o.w = w0 * v[3] + w1 * v[4] + w0 * v[5];
        o.x = o.x < 0.0f ? 0.0f : (o.x > 1.0f ? 1.0f : o.x);
        o.y = o.y < 0.0f ? 0.0f : (o.y > 1.0f ? 1.0f : o.y);
        o.z = o.z < 0.0f ? 0.0f : (o.z > 1.0f ? 1.0f : o.z);
        o.w = o.w < 0.0f ? 0.0f : (o.w > 1.0f ? 1.0f : o.w);
        *reinterpret_cast<float4*>(out + ch * HW + i * 1024 + j0) = o;
    }
}

extern "C" void kernel_launch(void* const* d_in, const int* in_sizes, int n_in,
                              void* d_out, int out_size, void* d_ws, size_t ws_size,
                              hipStream_t stream) {
    (void)in_sizes; (void)n_in; (void)out_size; (void)ws_size;
    const float* img = (const float*)d_in[0];
    const int2* dxdy = (const int2*)d_in[1];
    float* out = (float*)d_out;
    float* b1 = (float*)d_ws;  // 3 * 1024 * 1024 floats = 12 MB scratch

    // Gaussian weights, computed in f32 exactly like the reference.
    const float t = expf(-1.0f / (2.0f * 0.7f * 0.7f));
    const float sum = 1.0f + 2.0f * t;
    const float w0 = t / sum;    // side tap
    const float w1 = 1.0f / sum; // center tap

    k_blur1<<<dim3(32, 32, 3), dim3(256), 0, stream>>>(img, b1, w0, w1);
    k_glass2<<<dim3(32, 32), dim3(256), 0, stream>>>(b1, dxdy, out, w0, w1);
}